// AttentionEncoder_80118319939832
// MI455X (gfx1250) — compile-verified
//
#include <hip/hip_runtime.h>

typedef __attribute__((ext_vector_type(2))) float v2f;
typedef __attribute__((ext_vector_type(8))) float v8f;

// Problem constants (from reference)
#define Bsz 16
#define Uu 9
#define Nn 9
#define Ff 3750
#define Ll 300
#define K1 303750   // U * N * F   (K1 % 4 == 2 -> one partial step)
#define O1 2700     // U * L
#define K2 2700     // U * L
#define O2 81       // U * A * N
#define SPLITK 8
#define WAVES 8
#define KFULL 75937           // K1 / 4 : number of full 4-wide K steps
#define KSTEPS 75938          // ceil(K1 / 4)
#define STEPS_PER_UNIT 1187   // ceil(KSTEPS / (SPLITK*WAVES))

// ---------------------------------------------------------------------------
// Stage 1: h_partial[s] = x[16,K1] * W1^T tile, f32 WMMA 16x16x4, split-K.
// grid = (169 N-tiles, SPLITK), block = 256 (8 waves, each a K-subchunk),
// LDS reduction across the 8 waves, deterministic partial store.
// ---------------------------------------------------------------------------
__global__ __launch_bounds__(256) void gemm1_splitk(
    const float* __restrict__ x, const float* __restrict__ W1,
    float* __restrict__ partial)
{
    __shared__ float lds[256 * 8];
    const int tid  = threadIdx.x;
    const int lane = tid & 31;
    const int wave = tid >> 5;
    const int half = lane >> 4;    // 0: k0..k0+1 | 1: k0+2..k0+3
    const int lr   = lane & 15;
    const int oTile = blockIdx.x;
    const int split = blockIdx.y;

    const int o  = oTile * 16 + lr;            // output channel (B-matrix col)
    const int oc = (o < O1) ? o : (O1 - 1);    // clamp loads for last tile

    const size_t aRow = (size_t)lr * (size_t)K1;   // x row = batch m = lr
    const size_t bRow = (size_t)oc * (size_t)K1;   // W1 row = channel

    // K-unit of this wave; all derived values are wave-uniform, so pin them
    // to SGPRs with readfirstlane to get a clean scalar-branch loop.
    const int unit   = split * WAVES + wave;                 // 0..63
    const int sBegin = unit * STEPS_PER_UNIT;
    int sFullEnd = sBegin + STEPS_PER_UNIT;
    if (sFullEnd > KFULL) sFullEnd = KFULL;                  // exclude tail step
    int nFull = sFullEnd - sBegin;
    if (nFull < 0) nFull = 0;
    nFull = __builtin_amdgcn_readfirstlane(nFull);
    const int hasTail = __builtin_amdgcn_readfirstlane(unit == (SPLITK * WAVES - 1));

    const float* pA = x  + aRow + (size_t)sBegin * 4 + 2 * half;
    const float* pB = W1 + bRow + (size_t)sBegin * 4 + 2 * half;

    v8f c = {};
    #pragma unroll 8
    for (int i = 0; i < nFull; ++i) {
        v2f a = *(const v2f*)pA;
        v2f b = *(const v2f*)pB;
        c = __builtin_amdgcn_wmma_f32_16x16x4_f32(
                false, a, false, b, (short)0, c, false, false);
        pA += 4;
        pB += 4;
    }

    if (hasTail) {
        // Global tail step s = KFULL: k = 303748,303749 valid (half==0 lanes),
        // 303750,303751 OOB (half==1 lanes) -> feed zeros.
        const size_t ka = (size_t)KFULL * 4 + 2 * half;
        v2f a = {}, b = {};
        if (!half) {
            a = *(const v2f*)(x  + aRow + ka);
            b = *(const v2f*)(W1 + bRow + ka);
        }
        c = __builtin_amdgcn_wmma_f32_16x16x4_f32(
                false, a, false, b, (short)0, c, false, false);
    }

    #pragma unroll
    for (int r = 0; r < 8; ++r) lds[tid * 8 + r] = c[r];
    __syncthreads();

    if (tid < 32) {                            // wave 0 reduces the 8 tiles
        #pragma unroll
        for (int r = 0; r < 8; ++r) {
            float sacc = 0.0f;
            #pragma unroll
            for (int w = 0; w < WAVES; ++w) sacc += lds[(w * 32 + lane) * 8 + r];
            const int m = r + 8 * half;        // D row = batch index
            if (o < O1)
                partial[((size_t)split * Bsz + m) * O1 + o] = sacc;
        }
    }
}

// ---------------------------------------------------------------------------
// Stage 2: reduce split-K partials + bias + BatchNorm(inference) -> h[16,2700]
// ---------------------------------------------------------------------------
__global__ __launch_bounds__(256) void reduce_bn(
    const float* __restrict__ partial, const float* __restrict__ b1,
    const float* __restrict__ gamma, const float* __restrict__ beta,
    const float* __restrict__ rmean, const float* __restrict__ rvar,
    float* __restrict__ h)
{
    const int i = blockIdx.x * blockDim.x + threadIdx.x;
    if (i >= Bsz * O1) return;
    const int o = i % O1;
    float s = b1[o];
    #pragma unroll
    for (int sp = 0; sp < SPLITK; ++sp)
        s += partial[(size_t)sp * (Bsz * O1) + i];
    const float inv = rsqrtf(rvar[o] + 1e-5f);
    h[i] = gamma[o] * (s - rmean[o]) * inv + beta[o];
}

// ---------------------------------------------------------------------------
// Stage 3: logits[16,81] = h[16,2700] * W2^T + b2, f32 WMMA, 6 N-tiles.
// ---------------------------------------------------------------------------
__global__ __launch_bounds__(32) void gemm2(
    const float* __restrict__ h, const float* __restrict__ W2,
    const float* __restrict__ b2, float* __restrict__ logits)
{
    const int lane = threadIdx.x & 31;
    const int half = lane >> 4;
    const int lr   = lane & 15;
    const int o  = blockIdx.x * 16 + lr;
    const int oc = (o < O2) ? o : (O2 - 1);

    const float* pA = h  + (size_t)lr * K2 + 2 * half;
    const float* pB = W2 + (size_t)oc * K2 + 2 * half;

    v8f c = {};
    #pragma unroll 5
    for (int k0 = 0; k0 < K2; k0 += 4) {       // K2 % 4 == 0, no tail
        v2f a = *(const v2f*)pA;
        v2f b = *(const v2f*)pB;
        c = __builtin_amdgcn_wmma_f32_16x16x4_f32(
                false, a, false, b, (short)0, c, false, false);
        pA += 4;
        pB += 4;
    }
    #pragma unroll
    for (int r = 0; r < 8; ++r) {
        const int m = r + 8 * half;
        if (o < O2) logits[m * O2 + o] = c[r] + b2[o];
    }
}

// ---------------------------------------------------------------------------
// Stage 4: gumbel-softmax over N=9, then attention pooling over input.
// One block per (b,u); prob computed redundantly per thread (9 values).
// ---------------------------------------------------------------------------
__global__ __launch_bounds__(256) void softmax_pool(
    const float* __restrict__ logits, const float* __restrict__ gumbel,
    const float* __restrict__ tempPtr, const float* __restrict__ input,
    float* __restrict__ out)
{
    const int bu = blockIdx.x;          // 0 .. B*U-1
    const int b = bu / Uu, u = bu % Uu;
    const float invT = 1.0f / tempPtr[0];

    float z[Nn];
    float zmax = -3.4e38f;
    #pragma unroll
    for (int n = 0; n < Nn; ++n) {
        const float l = logits[b * O2 + u * Nn + n];
        const float g = gumbel[(b * Uu + u) * Nn + n];
        z[n] = (l + g) * invT;
        zmax = fmaxf(zmax, z[n]);
    }
    float denom = 0.0f;
    #pragma unroll
    for (int n = 0; n < Nn; ++n) { z[n] = __expf(z[n] - zmax); denom += z[n]; }
    const float inv = 1.0f / denom;
    #pragma unroll
    for (int n = 0; n < Nn; ++n) z[n] *= inv;

    const float* inBase  = input + (size_t)(b * Uu + u) * Nn * Ff;
    float*       outBase = out   + (size_t)(b * Uu + u) * Ff;
    for (int f = threadIdx.x; f < Ff; f += blockDim.x) {
        float acc = 0.0f;
        #pragma unroll
        for (int n = 0; n < Nn; ++n) acc += z[n] * inBase[(size_t)n * Ff + f];
        outBase[f] = acc;
    }
}

// ---------------------------------------------------------------------------
extern "C" void kernel_launch(void* const* d_in, const int* in_sizes, int n_in,
                              void* d_out, int out_size, void* d_ws, size_t ws_size,
                              hipStream_t stream)
{
    const float* input  = (const float*)d_in[0];
    const float* temp   = (const float*)d_in[1];
    const float* W1     = (const float*)d_in[2];
    const float* b1     = (const float*)d_in[3];
    const float* gamma  = (const float*)d_in[4];
    const float* beta   = (const float*)d_in[5];
    const float* rmean  = (const float*)d_in[6];
    const float* rvar   = (const float*)d_in[7];
    const float* W2     = (const float*)d_in[8];
    const float* b2     = (const float*)d_in[9];
    const float* gumbel = (const float*)d_in[10];
    float* out = (float*)d_out;

    float* wsP = (float*)d_ws;                       // SPLITK*16*2700 partials
    float* wsH = wsP + (size_t)SPLITK * Bsz * O1;    // 16*2700 h
    float* wsL = wsH + (size_t)Bsz * O1;             // 16*81 logits

    dim3 g1((O1 + 15) / 16, SPLITK);
    gemm1_splitk<<<g1, 256, 0, stream>>>(input, W1, wsP);

    reduce_bn<<<(Bsz * O1 + 255) / 256, 256, 0, stream>>>(
        wsP, b1, gamma, beta, rmean, rvar, wsH);

    gemm2<<<(O2 + 15) / 16, 32, 0, stream>>>(wsH, W2, b2, wsL);

    softmax_pool<<<Bsz * Uu, 256, 0, stream>>>(wsL, gumbel, temp, input, out);
}